// GCNLayer_13649406067044
// MI455X (gfx1250) — compile-verified
//
#include <hip/hip_runtime.h>

// GCN layer for MI455X (gfx1250, wave32).
//   N = 8192, D_IN = D_OUT = 512, A in {0,1}, avg degree ~33, symmetric, zero diag.
// Pipeline:
//   K1: dinv[i] = rsqrt(1 + rowsum(A[i]))                       (1 scan of A, 256MB)
//   K2: per-row ballot-compacted neighbor list in LDS,          (1 scan of A)
//       h[i] = dinv[i]*(dinv[i]*x[i] + sum_j dinv[j]*x[j])      (x stays in L2)
//   K3: out = relu(h @ W^T + b), f32 WMMA, double-buffered LDS panels filled
//       with GLOBAL_LOAD_ASYNC_TO_LDS_B128 (ASYNCcnt) when available.

typedef __attribute__((ext_vector_type(2))) float v2f;
typedef __attribute__((ext_vector_type(8))) float v8f;

#define NROWS 8192
#define DIM   512
#define NBR_CAP 1024

// ---- async-copy support (guarded; fallback = sync copy through regs) ----
#if __has_builtin(__builtin_amdgcn_global_load_async_to_lds_b128)
#define HAVE_ASYNC 1
#endif

#ifdef HAVE_ASYNC
#if __has_builtin(__builtin_amdgcn_s_wait_asynccnt)
#define WAIT_ASYNC(n) __builtin_amdgcn_s_wait_asynccnt(n)
#else
#define WAIT_ASYNC(n) asm volatile("s_wait_asynccnt %0" ::"i"(n) : "memory")
#endif
#else
#define WAIT_ASYNC(n)
#endif

#define AS1 __attribute__((address_space(1)))
#define AS3 __attribute__((address_space(3)))
typedef int v4i_ __attribute__((vector_size(16)));

__device__ __forceinline__ void cp_b128(const float* g, float* l) {
#ifdef HAVE_ASYNC
    __builtin_amdgcn_global_load_async_to_lds_b128(
        (AS1 v4i_*)(AS1 void*)(void*)(g),
        (AS3 v4i_*)(AS3 void*)(void*)(l), 0, 0);
#else
    *(float4*)l = *(const float4*)g;
#endif
}

// ---------------------------------------------------------------- kernel 1
__global__ __launch_bounds__(256) void k_degree(const float* __restrict__ A,
                                                float* __restrict__ dinv) {
    const int i = blockIdx.x;
    const int tid = threadIdx.x;
    const float4* row = (const float4*)(A + (size_t)i * NROWS);
    float s = 0.0f;
    for (int c = tid; c < NROWS / 4; c += 256) {
        float4 v = row[c];
        s += v.x + v.y + v.z + v.w;
    }
    __shared__ float red[256];
    red[tid] = s;
    __syncthreads();
    for (int st = 128; st > 0; st >>= 1) {
        if (tid < st) red[tid] += red[tid + st];
        __syncthreads();
    }
    if (tid == 0) dinv[i] = rsqrtf(red[0] + 1.0f);   // self-loop => deg >= 1
}

// ---------------------------------------------------------------- kernel 2
__global__ __launch_bounds__(256) void k_aggregate(const float* __restrict__ A,
                                                   const float* __restrict__ x,
                                                   const float* __restrict__ dinv,
                                                   float* __restrict__ h) {
    const int i   = blockIdx.x;
    const int tid = threadIdx.x;
    const int lane = tid & 31;
    const int wid  = tid >> 5;

    __shared__ int s_idx[NBR_CAP];
    __shared__ int warpCnt[8];

    // ---- deterministic ballot-compaction of nonzero columns of row i ----
    const float4* row = (const float4*)(A + (size_t)i * NROWS);
    int count = 0;
    const unsigned long long lanemask = (1ull << lane) - 1ull;
    for (int chunk = 0; chunk < NROWS / 4; chunk += 256) {   // 8 chunks of 1024 cols
        float4 v = row[chunk + tid];
        float vals[4] = {v.x, v.y, v.z, v.w};
        #pragma unroll
        for (int e = 0; e < 4; ++e) {
            const bool p = (vals[e] != 0.0f);
            const unsigned long long bal = __ballot(p);
            const int inwave = __popcll(bal & lanemask);
            if (lane == 0) warpCnt[wid] = __popcll(bal);
            __syncthreads();
            int off = 0, total = 0;
            #pragma unroll
            for (int w = 0; w < 8; ++w) {
                const int c = warpCnt[w];
                if (w < wid) off += c;
                total += c;
            }
            const int pos = count + off + inwave;
            if (p && pos < NBR_CAP) s_idx[pos] = (chunk + tid) * 4 + e;
            count += total;                 // same value in every thread
            __syncthreads();                // warpCnt reused next round
        }
    }
    if (count > NBR_CAP) count = NBR_CAP;

    // ---- gather-accumulate: 2 channels per thread, x rows L2-resident ----
    const float2* x2 = (const float2*)x;
    float2 acc = make_float2(0.0f, 0.0f);
    for (int n = 0; n < count; ++n) {
        const int j = s_idx[n];
        const float wj = dinv[j];
        const float2 xv = x2[(size_t)j * (DIM / 2) + tid];
        acc.x += wj * xv.x;
        acc.y += wj * xv.y;
    }
    const float di = dinv[i];
    const float2 xs = x2[(size_t)i * (DIM / 2) + tid];
    float2 hv;
    hv.x = di * (di * xs.x + acc.x);
    hv.y = di * (di * xs.y + acc.y);
    ((float2*)h)[(size_t)i * (DIM / 2) + tid] = hv;
}

// ---------------------------------------------------------------- kernel 3
// out[r][c] = relu( sum_k h[r][k] * W[c][k] + b[c] )
// Block: 256 thr / 8 waves, tile 128(M) x 64(N), K panels of 64,
// double-buffered LDS staging of both h and W panels.
// LDS row stride 68 floats => bank = (4*row + col) % 64: conflict-free frags.
#define BM 128
#define BN 64
#define BK 64
#define LSTR 68
#define CP_OPS 12   // async b128 instructions issued per panel (8 for h, 4 for W)

__global__ __launch_bounds__(256) void k_gemm_relu(const float* __restrict__ h,
                                                   const float* __restrict__ W,
                                                   const float* __restrict__ bias,
                                                   float* __restrict__ out) {
    __shared__ float shA[2][BM * LSTR];   // 128x64 panel of h   (69.6 KB total)
    __shared__ float shB[2][BN * LSTR];   //  64x64 panel of W   (34.8 KB total)

    const int tid  = threadIdx.x;
    const int lane = tid & 31;
    const int wid  = tid >> 5;
    const int mtile = blockIdx.x >> 3;           // 0..63
    const int ntile = blockIdx.x & 7;            // 0..7
    const int brow0 = mtile * BM;
    const int col0  = ntile * BN;

    const int m  = lane & 15;
    const int kh = lane >> 4;                    // 0 or 1 (K half)

    // panel-copy addressing: thread t covers (row = t>>4 + 16*it, col4 = (t&15)*4)
    const int cr = tid >> 4;
    const int cc = (tid & 15) * 4;

    // issue one K panel (h: 128 rows, W: 64 rows) into buffer `buf`
    auto copy_panel = [&](int buf, int k0) {
        #pragma unroll
        for (int it = 0; it < 8; ++it) {                       // h panel: 8 x b128
            const int r = cr + 16 * it;
            cp_b128(h + (size_t)(brow0 + r) * DIM + k0 + cc,
                    &shA[buf][r * LSTR + cc]);
        }
        #pragma unroll
        for (int it = 0; it < 4; ++it) {                       // W panel: 4 x b128
            const int r = cr + 16 * it;
            cp_b128(W + (size_t)(col0 + r) * DIM + k0 + cc,
                    &shB[buf][r * LSTR + cc]);
        }
    };

    v8f acc[4] = {v8f{}, v8f{}, v8f{}, v8f{}};

    copy_panel(0, 0);
    for (int kp = 0; kp < DIM / BK; ++kp) {
        const int cur = kp & 1;
        if (kp + 1 < DIM / BK) {
            copy_panel(cur ^ 1, (kp + 1) * BK);   // buffer free since barrier of kp-1
            WAIT_ASYNC(CP_OPS);                   // panel kp's copies have landed
        } else {
            WAIT_ASYNC(0);
        }
        __syncthreads();                          // all waves' copies visible

        const float* sA = &shA[cur][(wid * 16 + m) * LSTR];
        const float* sB = &shB[cur][m * LSTR];
        #pragma unroll 4
        for (int ks = 0; ks < BK / 4; ++ks) {
            const int kk = ks * 4 + 2 * kh;
            const v2f a  = *(const v2f*)(sA + kk);                 // ds_load_b64
            const v2f b0 = *(const v2f*)(sB + kk);
            const v2f b1 = *(const v2f*)(sB + 16 * LSTR + kk);
            const v2f b2 = *(const v2f*)(sB + 32 * LSTR + kk);
            const v2f b3 = *(const v2f*)(sB + 48 * LSTR + kk);
            acc[0] = __builtin_amdgcn_wmma_f32_16x16x4_f32(false, a, false, b0,
                                                           (short)0, acc[0], false, false);
            acc[1] = __builtin_amdgcn_wmma_f32_16x16x4_f32(false, a, false, b1,
                                                           (short)0, acc[1], false, false);
            acc[2] = __builtin_amdgcn_wmma_f32_16x16x4_f32(false, a, false, b2,
                                                           (short)0, acc[2], false, false);
            acc[3] = __builtin_amdgcn_wmma_f32_16x16x4_f32(false, a, false, b3,
                                                           (short)0, acc[3], false, false);
        }
        __syncthreads();                          // done with `cur` before refill
    }

    // C/D layout: VGPR v holds (row = v + 8*kh, col = lane&15).
    const int row0 = brow0 + wid * 16;
    #pragma unroll
    for (int t = 0; t < 4; ++t) {
        const int col = col0 + t * 16 + (lane & 15);
        const float bv = bias[col];
        #pragma unroll
        for (int v = 0; v < 8; ++v) {
            const int row = row0 + v + 8 * kh;
            const float r = acc[t][v] + bv;
            out[(size_t)row * DIM + col] = r > 0.0f ? r : 0.0f;
        }
    }
}

// ---------------------------------------------------------------- launcher
extern "C" void kernel_launch(void* const* d_in, const int* in_sizes, int n_in,
                              void* d_out, int out_size, void* d_ws, size_t ws_size,
                              hipStream_t stream) {
    const float* x = (const float*)d_in[0];   // [8192, 512]
    const float* A = (const float*)d_in[1];   // [8192, 8192]
    const float* W = (const float*)d_in[2];   // [512, 512]
    const float* b = (const float*)d_in[3];   // [512]
    float* out = (float*)d_out;               // [8192, 512]

    float* h    = (float*)d_ws;                                        // 16 MB
    float* dinv = (float*)((char*)d_ws + (size_t)NROWS * DIM * sizeof(float));

    k_degree<<<NROWS, 256, 0, stream>>>(A, dinv);
    k_aggregate<<<NROWS, 256, 0, stream>>>(A, x, dinv, h);
    k_gemm_relu<<<(NROWS / BM) * (DIM / BN), 256, 0, stream>>>(h, W, b, out);
}